// YoloV3Trainer_90890097918135
// MI455X (gfx1250) — compile-verified
//
#include <hip/hip_runtime.h>
#include <math.h>

typedef __attribute__((ext_vector_type(2))) float v2f;
typedef __attribute__((ext_vector_type(8))) float v8f;

#define BB 16
#define LL 50
#define NCLS 80
#define NA 3
#define NPRI 10647      // 507 + 2028 + 8112
#define OFF1 507
#define OFF2 2535
#define INPUTF 416.0f
#define NBLK 2048       // loss blocks per level
#define NSLOT (3*NBLK)

__constant__ int c_AW[9] = {116,156,373, 30,62,59, 10,16,33};
__constant__ int c_AH[9] = {90,198,326, 61,45,119, 13,30,23};

__device__ __forceinline__ void decode_prior(int n, int& lvl, int& S, int& i, int& j, int& a) {
    if (n < OFF1)      { lvl = 0; S = 13; }
    else if (n < OFF2) { lvl = 1; S = 26; n -= OFF1; }
    else               { lvl = 2; S = 52; n -= OFF2; }
    a = n % NA; int cell = n / NA; j = cell % S; i = cell / S;
}

// ---------------- init: prior table + clear scatter buffers ----------------
__global__ void k_init(float* __restrict__ priors, int* __restrict__ winner,
                       unsigned* __restrict__ supp, unsigned* __restrict__ clsmask) {
    const int total = 3 * BB * NPRI;
    for (int t = blockIdx.x * blockDim.x + threadIdx.x; t < total; t += gridDim.x * blockDim.x) {
        if (t < NPRI) {
            int lvl, S, i, j, a; decode_prior(t, lvl, S, i, j, a);
            float w = (float)c_AW[lvl*3+a] / INPUTF, h = (float)c_AH[lvl*3+a] / INPUTF;
            float cx = (j + 0.5f) / (float)S, cy = (i + 0.5f) / (float)S;
            priors[t*4+0] = cx - 0.5f*w; priors[t*4+1] = cy - 0.5f*h;
            priors[t*4+2] = w;           priors[t*4+3] = h;
        }
        if (t < BB * NPRI) { winner[t] = -1; supp[t] = 0u; }
        clsmask[t] = 0u;
    }
}

__device__ __forceinline__ float iou_of(const float* __restrict__ priors, int n,
                                        float bx, float by, float bw, float bh, float barea) {
    float px = priors[n*4+0], py = priors[n*4+1], pw = priors[n*4+2], ph = priors[n*4+3];
    float ix = fmaxf(px, bx), iy = fmaxf(py, by);
    float ax = fminf(px + pw, bx + bw), ay = fminf(py + ph, by + bh);
    float inter = fmaxf(ax - ix, 0.0f) * fmaxf(ay - iy, 0.0f);
    return inter / (pw*ph + barea - inter);
}

// ---------------- targets: one wave per label ----------------
__global__ void k_targets(const float* __restrict__ labels, const float* __restrict__ priors,
                          int* __restrict__ winner, unsigned* __restrict__ supp,
                          unsigned* __restrict__ clsmask, float* __restrict__ t_all) {
    int g = blockIdx.x;            // global label id = b*LL + l
    int b = g / LL;
    int lane = threadIdx.x;
    const float* lab = labels + (size_t)g * 5;
    float cls = lab[0];
    if (cls < 0.0f) return;        // invalid label: a complete no-op (uniform)
    float bx = lab[1], by = lab[2], bw = lab[3], bh = lab[4];
    float barea = bw * bh;

    // pass A: max IoU + suppression scatter
    float maxiou = -1.0f;
    for (int base = 0; base < NPRI; base += 32) {
        int n = base + lane;
        float iou = -1.0f;
        if (n < NPRI) {
            iou = iou_of(priors, n, bx, by, bw, bh, barea);
            if (iou >= 0.5f) atomicOr(&supp[(size_t)b*NPRI + n], 1u);
        }
        maxiou = fmaxf(maxiou, iou);
    }
    for (int d = 16; d > 0; d >>= 1) maxiou = fmaxf(maxiou, __shfl_xor(maxiou, d, 32));

    // pass B1: count exact ties (bitwise-identical recompute)
    int k = 0;
    for (int base = 0; base < NPRI; base += 32) {
        int n = base + lane;
        bool tie = (n < NPRI) && (iou_of(priors, n, bx, by, bw, bh, barea) == maxiou);
        k += __popcll(__ballot(tie));
    }
    int rank = (k - 1) / 2 + 1;    // median tie, matching reference

    // pass B2: find rank-th tied prior (ascending n)
    int idx = -1, running = 0;
    for (int base = 0; base < NPRI && idx < 0; base += 32) {
        int n = base + lane;
        bool tie = (n < NPRI) && (iou_of(priors, n, bx, by, bw, bh, barea) == maxiou);
        unsigned long long m = __ballot(tie);
        int kc = __popcll(m);
        if (running + kc >= rank) {
            int pos = rank - running;  // 1-based within this chunk
            unsigned long long below = m & ((1ull << lane) - 1ull);
            bool cand = tie && ((int)__popcll(below) + 1 == pos);
            unsigned long long cm = __ballot(cand);
            int src = __ffsll((unsigned long long)cm) - 1;
            idx = __shfl(n, src, 32);
        }
        running += kc;
    }

    if (lane == 0) {
        int lvl, S, i, j, a; decode_prior(idx, lvl, S, i, j, a);
        float pwc = (float)c_AW[lvl*3+a], phc = (float)c_AH[lvl*3+a];
        float dx = fminf(fmaxf((bx + 0.5f*bw) * (float)S - (float)j, 1e-6f), 0.999999f);
        float dy = fminf(fmaxf((by + 0.5f*bh) * (float)S - (float)i, 1e-6f), 0.999999f);
        float* t = t_all + (size_t)g * 4;
        t[0] = logf(dx / (1.0f - dx));
        t[1] = logf(dy / (1.0f - dy));
        t[2] = logf(bw * INPUTF / pwc);
        t[3] = logf(bh * INPUTF / phc);
        size_t key = (size_t)b * NPRI + idx;
        atomicMax(&winner[key], g);                       // segment_max of label id
        int ci = (int)cls;
        atomicOr(&clsmask[key*3 + (ci >> 5)], 1u << (ci & 31));
    }
}

// ---------------- loss streaming: one element per thread ----------------
__device__ __forceinline__ float bce_term(float x, float tt) {
    float p  = 1.0f / (1.0f + expf(-x));
    float lp  = fmaxf(logf(p),        -100.0f);
    float l1p = fmaxf(logf(1.0f - p), -100.0f);
    return -(tt * lp + (1.0f - tt) * l1p);
}

// S/LVLOFF compile-time: all index divisions strength-reduce to mulhi+shift.
template<int S, int LVLOFF>
__global__ void __launch_bounds__(256)
k_loss(const float* __restrict__ yout, const int* __restrict__ winner,
       const unsigned* __restrict__ supp, const unsigned* __restrict__ clsmask,
       const float* __restrict__ t_all, float* __restrict__ partials, int slotBase) {
    constexpr int SS2 = S * S;
    constexpr int CNT = BB * 255 * SS2;
    constexpr float invObj   = 1.0f / (float)(BB * NA * SS2);
    constexpr float invCoord = 1.0f / (float)(BB * NA * 4 * SS2);
    constexpr float invCls   = 1.0f / (float)(BB * NA * NCLS * SS2);

    float ao = 0.0f, ac = 0.0f, ak = 0.0f;
    for (int t = blockIdx.x * blockDim.x + threadIdx.x; t < CNT; t += gridDim.x * blockDim.x) {
        int b   = t / (255 * SS2);
        int rem = t - b * 255 * SS2;
        int ch  = rem / SS2;
        int pix = rem - ch * SS2;
        int i = pix / S, j = pix - i * S;
        int a = ch / 85, c = ch - a * 85;
        size_t key = (size_t)b * NPRI + (size_t)LVLOFF + (size_t)(i * S + j) * NA + a;
        int win = winner[key];
        bool chosen = (win >= 0);
        float x = yout[t];                       // coalesced: t is the memory index
        if (c == 4) {                            // objectness
            bool om = chosen || (supp[key] == 0u);
            if (om) ao += bce_term(x, chosen ? 1.0f : 0.0f) * invObj;
        } else if (c < 4) {                      // coords (MSE)
            if (chosen) {
                float d = x - t_all[(size_t)win * 4 + c];
                ac += d * d * invCoord;
            }
        } else {                                 // classes
            if (chosen) {
                int cc = c - 5;
                float tt = (float)((clsmask[key*3 + (cc >> 5)] >> (cc & 31)) & 1u);
                ak += bce_term(x, tt) * invCls;
            }
        }
    }
    __shared__ float sr[3 * 256];
    int tid = threadIdx.x;
    sr[tid] = ao; sr[256 + tid] = ac; sr[512 + tid] = ak;
    __syncthreads();
    for (int s = 128; s > 0; s >>= 1) {
        if (tid < s) {
            sr[tid]       += sr[tid + s];
            sr[256 + tid] += sr[256 + tid + s];
            sr[512 + tid] += sr[512 + tid + s];
        }
        __syncthreads();
    }
    if (tid == 0) {
        float* p = partials + (size_t)(slotBase + blockIdx.x) * 3;
        p[0] = sr[0]; p[1] = sr[256]; p[2] = sr[512];
    }
}

// ---------------- final reduction via V_WMMA_F32_16X16X4_F32 ----------------
// A (16x4 f32): lanes 0-15 hold K=0 (VGPR0=obj) and K=1 (VGPR1=coord) of row m=lane;
//               lanes 16-31 hold K=2 (cls) and K=3 (0) of row m=lane-16.
// B (4x16) = selection "identity" B[k][n] = (n==k): VGPR0 rows K0/K2 -> lanes 0,18;
//               VGPR1 rows K1/K3 -> lanes 1,19.
// D[m][n] accumulates A[m][n] for n<3 => columns 0/1/2 = obj/coord/cls sums per row.
__global__ void k_final(const float* __restrict__ partials, float* __restrict__ out, int P) {
    int lane = threadIdx.x;
    v2f bsel;
    bsel.x = (lane == 0 || lane == 18) ? 1.0f : 0.0f;
    bsel.y = (lane == 1 || lane == 19) ? 1.0f : 0.0f;
    v8f acc = {0.f,0.f,0.f,0.f,0.f,0.f,0.f,0.f};
    for (int base = 0; base < P; base += 16) {
        int i = base + (lane & 15);
        float o = 0.f, co = 0.f, cl = 0.f;
        if (i < P) { o = partials[i*3]; co = partials[i*3+1]; cl = partials[i*3+2]; }
        v2f a;
        a.x = (lane < 16) ? o  : cl;
        a.y = (lane < 16) ? co : 0.0f;
        acc = __builtin_amdgcn_wmma_f32_16x16x4_f32(
            /*neg_a=*/false, a, /*neg_b=*/false, bsel,
            /*c_mod=*/(short)0, acc, /*reuse_a=*/false, /*reuse_b=*/false);
    }
    // lane n (n<16) holds D[0..7][n]; lane 16+n holds D[8..15][n]
    float s = acc[0]+acc[1]+acc[2]+acc[3]+acc[4]+acc[5]+acc[6]+acc[7];
    __shared__ float red[32];
    red[lane] = s;
    __syncthreads();
    if (lane < 3) out[lane] = red[lane] + red[16 + lane];
}

extern "C" void kernel_launch(void* const* d_in, const int* in_sizes, int n_in,
                              void* d_out, int out_size, void* d_ws, size_t ws_size,
                              hipStream_t stream) {
    const float* y0     = (const float*)d_in[0];
    const float* y1     = (const float*)d_in[1];
    const float* y2     = (const float*)d_in[2];
    const float* labels = (const float*)d_in[3];
    float* out = (float*)d_out;

    // workspace layout (all 4-byte words; priors 16B-aligned at base)
    float*    priors  = (float*)d_ws;                       // NPRI*4
    int*      winner  = (int*)(priors + (size_t)NPRI * 4);  // BB*NPRI
    unsigned* supp    = (unsigned*)(winner + (size_t)BB * NPRI);   // BB*NPRI
    unsigned* clsmask = supp + (size_t)BB * NPRI;                  // 3*BB*NPRI
    float*    t_all   = (float*)(clsmask + (size_t)3 * BB * NPRI); // BB*LL*4
    float*    parts   = t_all + (size_t)BB * LL * 4;               // NSLOT*3

    int initTotal = 3 * BB * NPRI;
    k_init<<<(initTotal + 255) / 256, 256, 0, stream>>>(priors, winner, supp, clsmask);

    k_targets<<<BB * LL, 32, 0, stream>>>(labels, priors, winner, supp, clsmask, t_all);

    k_loss<13, 0>   <<<NBLK, 256, 0, stream>>>(y0, winner, supp, clsmask, t_all, parts, 0 * NBLK);
    k_loss<26, OFF1><<<NBLK, 256, 0, stream>>>(y1, winner, supp, clsmask, t_all, parts, 1 * NBLK);
    k_loss<52, OFF2><<<NBLK, 256, 0, stream>>>(y2, winner, supp, clsmask, t_all, parts, 2 * NBLK);

    k_final<<<1, 32, 0, stream>>>(parts, out, NSLOT);
}